// MultinomialLayer_18872086298693
// MI455X (gfx1250) — compile-verified
//
#include <hip/hip_runtime.h>
#include <stdint.h>

// ---------------------------------------------------------------------------
// Multinomial resampling for MI455X (gfx1250, wave32).
// Pipeline: zero -> block sums (WMMA f32 16x16x4) -> scan block sums ->
//           write cdf + coarse table -> sample (async-LDS table + 2-level
//           binary search + u32 atomics) -> count histogram -> threshold ->
//           deterministic counting-select compaction -> gather node rows.
// ---------------------------------------------------------------------------

typedef __attribute__((ext_vector_type(2))) float v2f;
typedef __attribute__((ext_vector_type(8))) float v8f;

__device__ __forceinline__ float u01_hash(unsigned x) {
  x ^= x >> 17; x *= 0xED5AD4BBu;
  x ^= x >> 11; x *= 0xAC4C1B51u;
  x ^= x >> 15; x *= 0x31848BABu;
  x ^= x >> 14;
  return (float)(x >> 8) * (1.0f / 16777216.0f);   // [0,1)
}

__device__ __forceinline__ v8f wmma_acc_sum(v2f a, v8f acc) {
  const v2f ones = {1.0f, 1.0f};
  return __builtin_amdgcn_wmma_f32_16x16x4_f32(
      /*neg_a=*/false, a, /*neg_b=*/false, ones,
      /*c_mod=*/(short)0, acc, /*reuse_a=*/false, /*reuse_b=*/false);
}

// ---- 0. zero accumulators (counts, hist, meta) ----------------------------
__global__ void k_zero(unsigned* __restrict__ counts, int N,
                       unsigned* __restrict__ hist, unsigned* __restrict__ meta) {
  unsigned gsize = gridDim.x * blockDim.x;
  unsigned tid = blockIdx.x * blockDim.x + threadIdx.x;
  for (unsigned i = tid; i < (unsigned)N; i += gsize) counts[i] = 0u;
  if (tid < 2048u) hist[tid] = 0u;
  if (tid < 8u) meta[tid] = 0u;
}

// ---- 1. per-block (4096 elem) sums via v_wmma_f32_16x16x4_f32 -------------
// Each wave feeds A = 16x4 f32 (2 VGPRs = 64 elems), B = ones. Every D entry
// replicates a row-sum, so sum(D over wave)/16 == sum of fed elements.
// Fast path: unconditional global_load_b64 per iteration (no lane guards).
__global__ void k_block_sums(const float* __restrict__ f, int N,
                             float* __restrict__ bsum) {
  __shared__ float red[256];
  const int tid = threadIdx.x;
  const int blockStart = blockIdx.x << 12;
  const int lane2 = ((tid >> 5) << 9) + ((tid & 31) << 1);  // wave*512 + lane*2
  v8f acc = {};
  if (blockStart + 4096 <= N) {
    const v2f* p = reinterpret_cast<const v2f*>(f + blockStart + lane2);
    #pragma unroll
    for (int it = 0; it < 8; ++it) {
      v2f a = p[it * 32];          // stride 32 v2f = 64 floats
      acc = wmma_acc_sum(a, acc);
    }
  } else {
    #pragma unroll
    for (int it = 0; it < 8; ++it) {
      int i = blockStart + lane2 + it * 64;
      v2f a;
      a.x = (i     < N) ? f[i]     : 0.0f;
      a.y = (i + 1 < N) ? f[i + 1] : 0.0f;
      acc = wmma_acc_sum(a, acc);
    }
  }
  float s = acc[0] + acc[1] + acc[2] + acc[3] + acc[4] + acc[5] + acc[6] + acc[7];
  red[tid] = s;
  __syncthreads();
  for (int off = 128; off > 0; off >>= 1) {
    if (tid < off) red[tid] += red[tid + off];
    __syncthreads();
  }
  if (tid == 0) bsum[blockIdx.x] = red[0] * 0.0625f;  // /16: D replicates row sums
}

// ---- 2. single-block scan of block sums (nBlocks <= 1024) -----------------
__global__ void k_scan_bsums(const float* __restrict__ bsum, int n,
                             float* __restrict__ boff, float* __restrict__ totalF) {
  __shared__ float s[1024];
  const int t = threadIdx.x;
  float v = (t < n) ? bsum[t] : 0.0f;
  s[t] = v;
  __syncthreads();
  for (int off = 1; off < 1024; off <<= 1) {
    float x = (t >= off) ? s[t - off] : 0.0f;
    __syncthreads();
    s[t] += x;
    __syncthreads();
  }
  if (t < n) boff[t] = s[t] - v;                 // exclusive prefix
  if (t == 0) totalF[0] = s[n > 0 ? n - 1 : 0];  // grand total (unnormalized)
}

// ---- 3. write unnormalized cdf + coarse table (every 1024th entry) --------
__global__ void k_write_cdf(const float* __restrict__ f, int N,
                            const float* __restrict__ boff,
                            float* __restrict__ cdf, float* __restrict__ tab) {
  __shared__ float s[256];
  const int t = threadIdx.x, b = blockIdx.x;
  const int blockStart = b << 12;
  const int base = blockStart + (t << 4);
  const bool full = (blockStart + 4096 <= N);
  float x[16];
  if (full) {
    const float4* p = reinterpret_cast<const float4*>(f + base);
    #pragma unroll
    for (int q = 0; q < 4; ++q) {
      float4 v = p[q];
      x[q * 4 + 0] = v.x; x[q * 4 + 1] = v.y;
      x[q * 4 + 2] = v.z; x[q * 4 + 3] = v.w;
    }
  } else {
    #pragma unroll
    for (int j = 0; j < 16; ++j) {
      int i = base + j;
      x[j] = (i < N) ? f[i] : 0.0f;
    }
  }
  float run = 0.0f;
  #pragma unroll
  for (int j = 0; j < 16; ++j) { run += x[j]; x[j] = run; }   // local inclusive
  s[t] = run;
  __syncthreads();
  const float v = run;
  for (int off = 1; off < 256; off <<= 1) {
    float y = (t >= off) ? s[t - off] : 0.0f;
    __syncthreads();
    s[t] += y;
    __syncthreads();
  }
  const float offs = boff[b] + (s[t] - v);
  if (full) {
    float4* q = reinterpret_cast<float4*>(cdf + base);
    #pragma unroll
    for (int g = 0; g < 4; ++g) {
      float4 v4;
      v4.x = offs + x[g * 4 + 0]; v4.y = offs + x[g * 4 + 1];
      v4.z = offs + x[g * 4 + 2]; v4.w = offs + x[g * 4 + 3];
      q[g] = v4;
    }
    // base is 16-aligned => (i & 1023)==1023 can only happen at j==15
    const int i15 = base + 15;
    if ((i15 & 1023) == 1023) tab[i15 >> 10] = offs + x[15];
  } else {
    #pragma unroll
    for (int j = 0; j < 16; ++j) {
      int i = base + j;
      if (i < N) {
        float c = offs + x[j];
        cdf[i] = c;
        if (((i & 1023) == 1023) || (i == N - 1)) tab[i >> 10] = c;
      }
    }
  }
}

// ---- 4. sampling: async-LDS coarse table + 2-level binary search ----------
__global__ void k_sample(const float* __restrict__ cdf, const float* __restrict__ tab,
                         int nTab, const float* __restrict__ totalF,
                         unsigned* __restrict__ counts, int N, unsigned totalCount) {
  __shared__ float sTab[4096];
  const int tid = threadIdx.x;
  // Broadcast coarse table HBM/L2 -> LDS with async b128 loads (ASYNCcnt path).
  const int nTab4 = (nTab + 3) >> 2;
  for (int j = tid; j < nTab4; j += 256) {
    unsigned ldsoff = (unsigned)(uintptr_t)(&sTab[j * 4]);
    const float* g = tab + (size_t)j * 4;
    asm volatile("global_load_async_to_lds_b128 %0, %1, off"
                 :: "v"(ldsoff), "v"(g) : "memory");
  }
  asm volatile("s_wait_asynccnt 0" ::: "memory");
  __syncthreads();

  const float total = totalF[0];
  const unsigned gsize = gridDim.x * blockDim.x;
  for (unsigned i = blockIdx.x * blockDim.x + tid; i < totalCount; i += gsize) {
    const float t = u01_hash(i) * total;
    // level 1: LDS coarse table -> 1024-wide segment
    int lo = 0, hi = nTab - 1;
    while (lo < hi) { int m = (lo + hi) >> 1; if (sTab[m] < t) lo = m + 1; else hi = m; }
    // level 2: L2-resident cdf segment
    int lo2 = lo << 10;
    int top = lo2 + 1024; if (top > N) top = N;
    int hi2 = top - 1;
    while (lo2 < hi2) { int m = (lo2 + hi2) >> 1; if (cdf[m] < t) lo2 = m + 1; else hi2 = m; }
    atomicAdd(&counts[lo2], 1u);
  }
}

// ---- 5. histogram of (clamped) counts -------------------------------------
__global__ void k_hist(const unsigned* __restrict__ counts, int N,
                       unsigned* __restrict__ hist) {
  __shared__ unsigned h[2048];
  const int tid = threadIdx.x;
  for (int j = tid; j < 2048; j += 256) h[j] = 0u;
  __syncthreads();
  const unsigned gsize = gridDim.x * blockDim.x;
  const unsigned gtid = blockIdx.x * blockDim.x + tid;
  const uint4* c4 = reinterpret_cast<const uint4*>(counts);
  const unsigned n4 = (unsigned)(N >> 2);
  for (unsigned i = gtid; i < n4; i += gsize) {
    uint4 v = c4[i];
    unsigned a = v.x > 2047u ? 2047u : v.x;
    unsigned b = v.y > 2047u ? 2047u : v.y;
    unsigned c = v.z > 2047u ? 2047u : v.z;
    unsigned d = v.w > 2047u ? 2047u : v.w;
    atomicAdd(&h[a], 1u); atomicAdd(&h[b], 1u);
    atomicAdd(&h[c], 1u); atomicAdd(&h[d], 1u);
  }
  for (unsigned i = (n4 << 2) + gtid; i < (unsigned)N; i += gsize) {
    unsigned c = counts[i];
    if (c > 2047u) c = 2047u;
    atomicAdd(&h[c], 1u);
  }
  __syncthreads();
  for (int j = tid; j < 2048; j += 256)
    if (h[j]) atomicAdd(&hist[j], h[j]);
}

// ---- 6. find top-k threshold T and tie budget -----------------------------
__global__ void k_threshold(const unsigned* __restrict__ hist,
                            const int* __restrict__ kptr, int N,
                            unsigned* __restrict__ meta) {
  if (threadIdx.x == 0 && blockIdx.x == 0) {
    unsigned k = (unsigned)kptr[0];
    if (k > (unsigned)N) k = (unsigned)N;
    unsigned acc = 0, T = 0, above = 0;
    for (int v = 2047; v >= 0; --v) {
      unsigned hv = hist[v];
      if (acc + hv >= k) { T = (unsigned)v; above = acc; break; }
      acc += hv;
    }
    meta[0] = T; meta[1] = k - above; meta[2] = 0u; meta[3] = k;
  }
}

// helper: load 16 counts (clamped), full-chunk path uses uint4 loads
__device__ __forceinline__ void load_counts16(const unsigned* __restrict__ counts,
                                              int base, int N, bool full,
                                              unsigned c[16], unsigned valid[16]) {
  if (full) {
    const uint4* p = reinterpret_cast<const uint4*>(counts + base);
    #pragma unroll
    for (int q = 0; q < 4; ++q) {
      uint4 v = p[q];
      c[q * 4 + 0] = v.x; c[q * 4 + 1] = v.y;
      c[q * 4 + 2] = v.z; c[q * 4 + 3] = v.w;
    }
    #pragma unroll
    for (int j = 0; j < 16; ++j) {
      if (c[j] > 2047u) c[j] = 2047u;
      valid[j] = 1u;
    }
  } else {
    #pragma unroll
    for (int j = 0; j < 16; ++j) {
      int i = base + j;
      valid[j] = (i < N) ? 1u : 0u;
      unsigned v = valid[j] ? counts[i] : 0u;
      c[j] = v > 2047u ? 2047u : v;
    }
  }
}

// ---- 7. per-block counts of (>T) and (==T) --------------------------------
__global__ void k_selcount(const unsigned* __restrict__ counts, int N,
                           const unsigned* __restrict__ meta,
                           unsigned* __restrict__ nA, unsigned* __restrict__ nE) {
  __shared__ unsigned sa[256], se[256];
  const int t = threadIdx.x, b = blockIdx.x;
  const unsigned T = meta[0];
  const int blockStart = b << 12;
  const int base = blockStart + (t << 4);
  const bool full = (blockStart + 4096 <= N);
  unsigned c[16], valid[16];
  load_counts16(counts, base, N, full, c, valid);
  unsigned a = 0, e = 0;
  #pragma unroll
  for (int j = 0; j < 16; ++j) {
    if (valid[j]) { a += (c[j] > T); e += (c[j] == T); }
  }
  sa[t] = a; se[t] = e;
  __syncthreads();
  for (int off = 128; off > 0; off >>= 1) {
    if (t < off) { sa[t] += sa[t + off]; se[t] += se[t + off]; }
    __syncthreads();
  }
  if (t == 0) { nA[b] = sa[0]; nE[b] = se[0]; }
}

// ---- 8. single-block exclusive scan of block select counts ----------------
__global__ void k_selscan(const unsigned* __restrict__ nA, const unsigned* __restrict__ nE,
                          int n, unsigned* __restrict__ aOff, unsigned* __restrict__ eOff,
                          unsigned* __restrict__ meta) {
  __shared__ unsigned sa[1024], se[1024];
  const int t = threadIdx.x;
  unsigned va = (t < n) ? nA[t] : 0u;
  unsigned ve = (t < n) ? nE[t] : 0u;
  sa[t] = va; se[t] = ve;
  __syncthreads();
  for (int off = 1; off < 1024; off <<= 1) {
    unsigned xa = (t >= off) ? sa[t - off] : 0u;
    unsigned xe = (t >= off) ? se[t - off] : 0u;
    __syncthreads();
    sa[t] += xa; se[t] += xe;
    __syncthreads();
  }
  if (t < n) { aOff[t] = sa[t] - va; eOff[t] = se[t] - ve; }
  if (t == 0) meta[2] = sa[n > 0 ? n - 1 : 0];   // totalAbove
}

// ---- 9. deterministic compaction + gather of node rows --------------------
__global__ void k_gather(const unsigned* __restrict__ counts,
                         const float* __restrict__ nodes, int N,
                         const unsigned* __restrict__ meta,
                         const unsigned* __restrict__ aOff,
                         const unsigned* __restrict__ eOff,
                         float* __restrict__ out) {
  __shared__ unsigned sa[256], se[256];
  const int t = threadIdx.x, b = blockIdx.x;
  const unsigned T = meta[0], ties = meta[1], totA = meta[2], k = meta[3];
  const int blockStart = b << 12;
  const int base = blockStart + (t << 4);
  const bool full = (blockStart + 4096 <= N);
  unsigned c[16], valid[16];
  load_counts16(counts, base, N, full, c, valid);
  unsigned aMask = 0, eMask = 0, a = 0, e = 0;
  #pragma unroll
  for (int j = 0; j < 16; ++j) {
    if (valid[j]) {
      if (c[j] > T)       { aMask |= (1u << j); ++a; }
      else if (c[j] == T) { eMask |= (1u << j); ++e; }
    }
  }
  sa[t] = a; se[t] = e;
  __syncthreads();
  const unsigned va = a, ve = e;
  for (int off = 1; off < 256; off <<= 1) {
    unsigned xa = (t >= off) ? sa[t - off] : 0u;
    unsigned xe = (t >= off) ? se[t - off] : 0u;
    __syncthreads();
    sa[t] += xa; se[t] += xe;
    __syncthreads();
  }
  const unsigned aBase = aOff[b] + (sa[t] - va);
  const unsigned eBase = eOff[b] + (se[t] - ve);
  unsigned ra = 0, re = 0;
  #pragma unroll
  for (int j = 0; j < 16; ++j) {
    int i = base + j;
    unsigned pos = 0xFFFFFFFFu;
    if (aMask & (1u << j)) {
      pos = aBase + ra++;
    } else if (eMask & (1u << j)) {
      unsigned r = eBase + re++;
      if (r < ties) pos = totA + r;
    }
    if (pos < k) {
      out[(size_t)pos * 3 + 0] = nodes[(size_t)i * 3 + 0];
      out[(size_t)pos * 3 + 1] = nodes[(size_t)i * 3 + 1];
      out[(size_t)pos * 3 + 2] = nodes[(size_t)i * 3 + 2];
    }
  }
}

// ---------------------------------------------------------------------------
extern "C" void kernel_launch(void* const* d_in, const int* in_sizes, int n_in,
                              void* d_out, int out_size, void* d_ws, size_t ws_size,
                              hipStream_t stream) {
  const float* f     = (const float*)d_in[0];
  const float* nodes = (const float*)d_in[1];
  const int*   kptr  = (const int*)d_in[2];
  float* out = (float*)d_out;
  const int N = in_sizes[0];
  (void)n_in; (void)out_size; (void)ws_size;

  char* w = (char*)d_ws;
  auto al = [](size_t x) { return (x + 255) & ~(size_t)255; };
  size_t o = 0;
  float*    cdf    = (float*)(w + o);    o = al(o + (size_t)N * 4);   // 16 MB
  unsigned* counts = (unsigned*)(w + o); o = al(o + (size_t)N * 4);   // 16 MB
  float*    tab    = (float*)(w + o);    o = al(o + 4096 * 4);
  float*    bsum   = (float*)(w + o);    o = al(o + 1024 * 4);
  float*    boff   = (float*)(w + o);    o = al(o + 1024 * 4);
  float*    totalF = (float*)(w + o);    o = al(o + 16);
  unsigned* meta   = (unsigned*)(w + o); o = al(o + 64);
  unsigned* hist   = (unsigned*)(w + o); o = al(o + 2048 * 4);
  unsigned* nA     = (unsigned*)(w + o); o = al(o + 1024 * 4);
  unsigned* nE     = (unsigned*)(w + o); o = al(o + 1024 * 4);
  unsigned* aOff   = (unsigned*)(w + o); o = al(o + 1024 * 4);
  unsigned* eOff   = (unsigned*)(w + o); o = al(o + 1024 * 4);

  const int nScan = (N + 4095) / 4096;           // <= 1024 for N <= 4.19M
  const int nTab  = (N + 1023) / 1024;           // <= 4096
  const unsigned totalCount = 10u * (unsigned)N; // reference: 10*N draws

  k_zero      <<<2048,  256, 0, stream>>>(counts, N, hist, meta);
  k_block_sums<<<nScan, 256, 0, stream>>>(f, N, bsum);
  k_scan_bsums<<<1,    1024, 0, stream>>>(bsum, nScan, boff, totalF);
  k_write_cdf <<<nScan, 256, 0, stream>>>(f, N, boff, cdf, tab);
  k_sample    <<<1024,  256, 0, stream>>>(cdf, tab, nTab, totalF, counts, N, totalCount);
  k_hist      <<<512,   256, 0, stream>>>(counts, N, hist);
  k_threshold <<<1,      64, 0, stream>>>(hist, kptr, N, meta);
  k_selcount  <<<nScan, 256, 0, stream>>>(counts, N, meta, nA, nE);
  k_selscan   <<<1,    1024, 0, stream>>>(nA, nE, nScan, aOff, eOff, meta);
  k_gather    <<<nScan, 256, 0, stream>>>(counts, nodes, N, meta, aOff, eOff, out);
}